// AmazonNet_36704790511894
// MI455X (gfx1250) — compile-verified
//
#include <hip/hip_runtime.h>

#define N_NODES   100000
#define N_EDGES   1600000
#define E_TOTAL   (N_EDGES + N_NODES)   // 1,700,000 (self loops appended)
#define IN_DIM    256
#define HID       64
#define HEADS     3
#define F_OUT     192                   // HEADS*HID
#define NUM_CLASSES 10
#define NUM_GRAPHS  64

typedef __attribute__((ext_vector_type(2))) float v2f;
typedef __attribute__((ext_vector_type(8))) float v8f;

__device__ __forceinline__ float lrelu(float v, float s) { return v > 0.0f ? v : v * s; }

__device__ __forceinline__ void atomic_max_f32(float* p, float v) {
  __hip_atomic_fetch_max(p, v, __ATOMIC_RELAXED, __HIP_MEMORY_SCOPE_AGENT);
}

// ---------------------------------------------------------------------------
// Workspace / output init
// ---------------------------------------------------------------------------
__global__ void init_ws_kernel(float* __restrict__ fc_acc, float* __restrict__ mbuf,
                               float* __restrict__ denom, float* __restrict__ psum,
                               float* __restrict__ cnt) {
  int t = blockIdx.x * blockDim.x + threadIdx.x;
  if (t < N_NODES * F_OUT) fc_acc[t] = 0.0f;
  if (t < N_NODES * HEADS) { mbuf[t] = -3.0e38f; denom[t] = 0.0f; }
  if (t < NUM_GRAPHS * F_OUT) psum[t] = 0.0f;
  if (t < NUM_GRAPHS) cnt[t] = 0.0f;
}

// ---------------------------------------------------------------------------
// xl = x @ W_l + b_l ; xr = x @ W_r + b_r  via V_WMMA_F32_16X16X4_F32
// One wave computes one 16x16 output tile. 6250 row blocks x 24 col tiles
// (12 for W_l, 12 for W_r) = 150,000 waves, exact division -> EXEC all ones.
// A frag (16x4 f32): lane l holds row m=l&15, K = (l>>4)*2 + {0,1}
// B frag (4x16 f32): lane l holds col n=l&15, K = (l>>4)*2 + {0,1}
// C/D (16x16 f32):   lane l vgpr j -> row m = (l>>4)*8 + j, col n = l&15
// ---------------------------------------------------------------------------
__global__ void __launch_bounds__(128)
gat_gemm_wmma(const float* __restrict__ x,
              const float* __restrict__ Wl, const float* __restrict__ bl,
              const float* __restrict__ Wr, const float* __restrict__ br,
              float* __restrict__ xl, float* __restrict__ xr) {
  const int wave = (blockIdx.x * 128 + threadIdx.x) >> 5;
  const int lane = threadIdx.x & 31;
  const int colTile = wave % 24;        // consecutive waves share the same x rows
  const int rowBlk  = wave / 24;
  const bool right = colTile >= 12;
  const int c0 = (right ? colTile - 12 : colTile) * 16;
  const float* __restrict__ W  = right ? Wr : Wl;
  const float* __restrict__ bb = right ? br : bl;
  float* __restrict__ o        = right ? xr : xl;
  const int r0 = rowBlk * 16;
  const int ln = lane & 15;
  const int hi = lane >> 4;             // 0 or 1

  const float* xrow = x + (size_t)(r0 + ln) * IN_DIM + hi * 2;
  const float* wcol = W + (size_t)(hi * 2) * F_OUT + c0 + ln;

  v8f acc = {};
  #pragma unroll 4
  for (int k = 0; k < IN_DIM; k += 4) {
    v2f a = *(const v2f*)(xrow + k);                 // x[r0+ln][k + hi*2 .. +1]
    v2f b;
    b.x = wcol[(size_t)k * F_OUT];                   // W[k + hi*2    ][c0+ln]
    b.y = wcol[(size_t)(k + 1) * F_OUT];             // W[k + hi*2 + 1][c0+ln]
    acc = __builtin_amdgcn_wmma_f32_16x16x4_f32(
        /*neg_a=*/false, a, /*neg_b=*/false, b,
        /*c_mod=*/(short)0, acc, /*reuse_a=*/false, /*reuse_b=*/false);
  }
  const float bv = bb[c0 + ln];
  #pragma unroll
  for (int j = 0; j < 8; ++j) {
    int m = hi * 8 + j;
    o[(size_t)(r0 + m) * F_OUT + c0 + ln] = acc[j] + bv;
  }
}

// ---------------------------------------------------------------------------
// Pass 1 over edges: logits[e,h] = sum_c att[h,c]*lrelu(xl[src]+xr[dst], 0.2)
// One wave per edge; segment max into mbuf via float atomic max.
// ---------------------------------------------------------------------------
__global__ void __launch_bounds__(256)
edge_logits_kernel(const int* __restrict__ ei,
                   const float* __restrict__ xl, const float* __restrict__ xr,
                   const float* __restrict__ att,
                   float* __restrict__ elog, float* __restrict__ mbuf) {
  const int wv = (blockIdx.x * 256 + threadIdx.x) >> 5;
  if (wv >= E_TOTAL) return;            // uniform per wave
  const int lane = threadIdx.x & 31;
  int src, dst;
  if (wv < N_EDGES) { src = ei[wv]; dst = ei[N_EDGES + wv]; }
  else { src = wv - N_EDGES; dst = src; }   // self loop
  const float* pl = xl + (size_t)src * F_OUT;
  const float* pr = xr + (size_t)dst * F_OUT;
  float s[HEADS];
  #pragma unroll
  for (int h = 0; h < HEADS; ++h) {
    int b = h * HID;
    float v0 = lrelu(pl[b + lane]      + pr[b + lane],      0.2f);
    float v1 = lrelu(pl[b + lane + 32] + pr[b + lane + 32], 0.2f);
    s[h] = v0 * att[b + lane] + v1 * att[b + lane + 32];
  }
  #pragma unroll
  for (int off = 16; off > 0; off >>= 1) {
    #pragma unroll
    for (int h = 0; h < HEADS; ++h) s[h] += __shfl_down(s[h], off, 32);
  }
  if (lane == 0) {
    #pragma unroll
    for (int h = 0; h < HEADS; ++h) {
      elog[(size_t)wv * HEADS + h] = s[h];
      atomic_max_f32(&mbuf[(size_t)dst * HEADS + h], s[h]);
    }
  }
}

// ---------------------------------------------------------------------------
// Pass 2: a = exp(logit - m[dst]); overwrite elog; denom[dst] += a
// ---------------------------------------------------------------------------
__global__ void edge_exp_kernel(const int* __restrict__ ei, float* __restrict__ elog,
                                const float* __restrict__ mbuf, float* __restrict__ denom) {
  int t = blockIdx.x * blockDim.x + threadIdx.x;
  if (t >= E_TOTAL * HEADS) return;
  int e = t / HEADS, h = t - e * HEADS;
  int dst = (e < N_EDGES) ? ei[N_EDGES + e] : (e - N_EDGES);
  float a = expf(elog[t] - mbuf[dst * HEADS + h]);
  elog[t] = a;
  atomicAdd(&denom[dst * HEADS + h], a);
}

// ---------------------------------------------------------------------------
// Pass 3: fc_acc[dst] += (a/denom[dst]) * xl[src]  (192 atomic f32 adds/edge,
// L2-resident since xl + fc fit comfortably in the 192MB L2)
// ---------------------------------------------------------------------------
__global__ void __launch_bounds__(256)
edge_accum_kernel(const int* __restrict__ ei, const float* __restrict__ elog,
                  const float* __restrict__ denom, const float* __restrict__ xl,
                  float* __restrict__ fc_acc) {
  const int wv = (blockIdx.x * 256 + threadIdx.x) >> 5;
  if (wv >= E_TOTAL) return;
  const int lane = threadIdx.x & 31;
  int src, dst;
  if (wv < N_EDGES) { src = ei[wv]; dst = ei[N_EDGES + wv]; }
  else { src = wv - N_EDGES; dst = src; }
  const float* pl = xl + (size_t)src * F_OUT;
  float* pa = fc_acc + (size_t)dst * F_OUT;
  #pragma unroll
  for (int h = 0; h < HEADS; ++h) {
    float alpha = elog[(size_t)wv * HEADS + h] / denom[(size_t)dst * HEADS + h];
    int b = h * HID;
    atomicAdd(&pa[b + lane],      alpha * pl[b + lane]);
    atomicAdd(&pa[b + lane + 32], alpha * pl[b + lane + 32]);
  }
}

// ---------------------------------------------------------------------------
// Node finalize: first_conv = acc + bias; pre_pool = lrelu(fc, 0.01);
// pooled sums + counts per graph.
// ---------------------------------------------------------------------------
__global__ void node_final_kernel(float* __restrict__ fc, const float* __restrict__ bias,
                                  const int* __restrict__ batch,
                                  float* __restrict__ prepool,
                                  float* __restrict__ psum, float* __restrict__ cnt) {
  int t = blockIdx.x * blockDim.x + threadIdx.x;
  if (t >= N_NODES * F_OUT) return;
  int i = t / F_OUT;
  int j = t - i * F_OUT;
  float v = fc[t] + bias[j];
  fc[t] = v;                                // first_conv output (in place)
  float pp = v > 0.0f ? v : 0.01f * v;
  prepool[t] = pp;
  int g = batch[i];
  atomicAdd(&psum[g * F_OUT + j], pp);
  if (j == 0) atomicAdd(&cnt[g], 1.0f);
}

// ---------------------------------------------------------------------------
// Mean pool + classifier (tiny): one block per graph.
// ---------------------------------------------------------------------------
__global__ void pool_cls_kernel(const float* __restrict__ psum, const float* __restrict__ cnt,
                                const float* __restrict__ Wc, const float* __restrict__ bc,
                                float* __restrict__ postpool, float* __restrict__ logits) {
  __shared__ float pp[F_OUT];
  int g = blockIdx.x, j = threadIdx.x;
  float c = cnt[g]; c = c < 1.0f ? 1.0f : c;
  float v = psum[g * F_OUT + j] / c;
  postpool[g * F_OUT + j] = v;
  pp[j] = v;
  __syncthreads();
  if (j < NUM_CLASSES) {
    float s = bc[j];
    #pragma unroll 4
    for (int k = 0; k < F_OUT; ++k) s += pp[k] * Wc[k * NUM_CLASSES + j];
    logits[g * NUM_CLASSES + j] = s;
  }
}

// ---------------------------------------------------------------------------
extern "C" void kernel_launch(void* const* d_in, const int* in_sizes, int n_in,
                              void* d_out, int out_size, void* d_ws, size_t ws_size,
                              hipStream_t stream) {
  (void)in_sizes; (void)n_in; (void)out_size; (void)ws_size;
  const float* x     = (const float*)d_in[0];
  const int*   ei    = (const int*)d_in[1];
  const int*   batch = (const int*)d_in[2];
  const float* Wl    = (const float*)d_in[3];
  const float* bl    = (const float*)d_in[4];
  const float* Wr    = (const float*)d_in[5];
  const float* br    = (const float*)d_in[6];
  const float* att   = (const float*)d_in[7];
  const float* bias  = (const float*)d_in[8];
  const float* Wc    = (const float*)d_in[9];
  const float* bc    = (const float*)d_in[10];

  // Output layout: logits[64*10] | pre_pool[100000*192] | post_pool[64*192] | first_conv[100000*192]
  float* out          = (float*)d_out;
  float* out_logits   = out;
  float* out_prepool  = out + 640;
  float* out_postpool = out + 640 + (size_t)N_NODES * F_OUT;
  float* out_fc       = out + 640 + (size_t)N_NODES * F_OUT + (size_t)NUM_GRAPHS * F_OUT;

  // Workspace layout (floats)
  float* ws    = (float*)d_ws;
  float* xl    = ws;                                   // 19,200,000
  float* xr    = xl   + (size_t)N_NODES * F_OUT;       // 19,200,000
  float* elog  = xr   + (size_t)N_NODES * F_OUT;       //  5,100,000
  float* mbuf  = elog + (size_t)E_TOTAL * HEADS;       //    300,000
  float* denom = mbuf + (size_t)N_NODES * HEADS;       //    300,000
  float* psum  = denom+ (size_t)N_NODES * HEADS;       //     12,288
  float* cnt   = psum + (size_t)NUM_GRAPHS * F_OUT;    //         64

  // 1) init accumulators
  init_ws_kernel<<<(N_NODES * F_OUT + 255) / 256, 256, 0, stream>>>(out_fc, mbuf, denom, psum, cnt);

  // 2) WMMA GEMMs: 6250*24 = 150,000 waves, 4 waves/block
  gat_gemm_wmma<<<37500, 128, 0, stream>>>(x, Wl, bl, Wr, br, xl, xr);

  // 3) edge logits + segment max (1.7M waves, 8/block)
  edge_logits_kernel<<<E_TOTAL / 8, 256, 0, stream>>>(ei, xl, xr, att, elog, mbuf);

  // 4) exp + segment sum
  edge_exp_kernel<<<(E_TOTAL * HEADS + 255) / 256, 256, 0, stream>>>(ei, elog, mbuf, denom);

  // 5) alpha-weighted scatter into first_conv accumulator
  edge_accum_kernel<<<E_TOTAL / 8, 256, 0, stream>>>(ei, elog, denom, xl, out_fc);

  // 6) bias + leaky_relu + per-graph pooled sums
  node_final_kernel<<<(N_NODES * F_OUT + 255) / 256, 256, 0, stream>>>(
      out_fc, bias, batch, out_prepool, psum, cnt);

  // 7) mean pool + classifier
  pool_cls_kernel<<<NUM_GRAPHS, F_OUT, 0, stream>>>(psum, cnt, Wc, bc, out_postpool, out_logits);
}